// EquivariantVelocityHead_89979564852029
// MI455X (gfx1250) — compile-verified
//
#include <hip/hip_runtime.h>

// EquivariantVelocityHead for MI455X (gfx1250, wave32).
// Phase 1: [A | Bm+b1] = h @ [Wa | Wb] via V_WMMA_F32_16X16X4_F32 (f32 matrix pipe, full
//          reference precision). One-wave workgroups -> tile indices are SALU-scalar, so
//          the A/B store split is a scalar branch (no exec-mask juggling).
// Phase 2: pairwise silu interaction. The per-batch A tile (256x128 f32) is staged into LDS
//          by the Tensor Data Mover (tensor_load_to_lds) using the D# LDS-padding feature
//          (pad after every 128 dwords, +1 dword) to produce a stride-129 layout:
//          bank = (j + h) % 64 -> conflict-free indexed reads. v_exp_f32 + v_rcp_f32 silu,
//          8 i-rows per thread, deterministic shfl_xor + fixed-order cross-wave reduction.

#define BB 8
#define NN 256
#define HH 128

typedef float v2f __attribute__((ext_vector_type(2)));
typedef float v8f __attribute__((ext_vector_type(8)));
typedef unsigned int v4u __attribute__((ext_vector_type(4)));
typedef int v4i __attribute__((ext_vector_type(4)));
typedef int v8i __attribute__((ext_vector_type(8)));

#if defined(__has_builtin)
#if __has_builtin(__builtin_amdgcn_tensor_load_to_lds) && \
    __has_builtin(__builtin_amdgcn_s_wait_tensorcnt)
#define EVH_USE_TDM 1
#endif
#endif

// ---------------- Phase 1: WMMA f32 GEMM ----------------
__global__ void __launch_bounds__(32)
evh_gemm_wmma(const float* __restrict__ hmat, const float* __restrict__ W1,
              const float* __restrict__ b1, float* __restrict__ Aout,
              float* __restrict__ Bout) {
  const int tileM = blockIdx.x >> 4;   // scalar: 128 tiles of 16 rows (M = 2048)
  const int tileN = blockIdx.x & 15;   // scalar: 16 tiles of 16 cols (N = 256 = [A | Bm])
  const int lane  = threadIdx.x;       // one wave per workgroup
  const int half  = lane >> 4;         // K-pair select per ISA 16x4 f32 A layout
  const int l     = lane & 15;
  const int m = tileM * 16 + l;
  const int n = tileN * 16 + l;
  // B[k][n] of W' = [Wa | Wb]:  n<128 -> W1[k][n],  else W1[128+k][n-128]
  const float* wcol = (tileN < 8) ? (W1 + n) : (W1 + 128 * HH + (n - 128));
  v8f acc = {};
  #pragma unroll 4
  for (int k = 0; k < HH; k += 4) {
    const int ka = k + half * 2;
    v2f a, bv;
    a.x  = hmat[m * HH + ka];
    a.y  = hmat[m * HH + ka + 1];
    bv.x = wcol[ka * HH];
    bv.y = wcol[(ka + 1) * HH];
    acc = __builtin_amdgcn_wmma_f32_16x16x4_f32(false, a, false, bv, (short)0, acc,
                                                false, false);
  }
  if (tileN < 8) {                      // scalar branch
    #pragma unroll
    for (int v = 0; v < 8; ++v) {
      const int row = tileM * 16 + half * 8 + v;   // C layout: VGPR v -> M = v + 8*half
      Aout[row * HH + tileN * 16 + l] = acc[v];
    }
  } else {
    #pragma unroll
    for (int v = 0; v < 8; ++v) {
      const int row = tileM * 16 + half * 8 + v;
      const int col = tileN * 16 + l - 128;
      Bout[row * HH + col] = acc[v] + b1[col];     // fold b1 into Bm
    }
  }
}

// ---------------- Phase 2: pairwise silu + reduction ----------------
__global__ void __launch_bounds__(256)
evh_pair_kernel(const float* __restrict__ pos, const float* __restrict__ Aout,
                const float* __restrict__ Bout, const float* __restrict__ W1,
                const float* __restrict__ W2, const float* __restrict__ b2,
                float* __restrict__ out) {
  constexpr int IBLK = 8;
  constexpr int ASTR = HH + 1;               // 129: (j*129+h)%64 = (j+h)%64, conflict-free
  extern __shared__ float smem[];            // dynamic LDS base = group offset 0
  float* As  = smem;                         // 256*129  (TDM-padded A tile)
  float* Bs  = As + NN * ASTR;               // 8*128   (Bm + b1 rows for this i-block)
  float* wds = Bs + IBLK * HH;               // 128     (wd = W1 row 2H)
  float* w2s = wds + HH;                     // 128
  float* pis = w2s + HH;                     // 8*3     (pos of i-block rows)
  float* red = pis + IBLK * 3;               // 8 waves * 24 partials

  const int tid = threadIdx.x;
  const int b   = blockIdx.x / (NN / IBLK);
  const int i0  = (blockIdx.x % (NN / IBLK)) * IBLK;
  const int rowBase = b * NN;

  // ---- stage the 256x128 A tile into LDS ----
#ifdef EVH_USE_TDM
  if (tid < 32) {                            // wave 0 issues the TDM descriptor
    const unsigned long long gaddr =
        (unsigned long long)(uintptr_t)(Aout + (size_t)rowBase * HH);
    // D# group 0: count=1 | lds_addr=0 | global_addr | type=2
    v4u g0 = { 1u,
               0u,
               (unsigned)(gaddr & 0xffffffffu),
               (unsigned)((gaddr >> 32) & 0x01ffffffu) | (2u << 30) };
    // D# group 1: data_size=4B, pad_enable, pad_interval=6 (128 dw), pad_amount=0 (+1 dw)
    //             tensor_dim0=128, tensor_dim1=256, tile_dim0=128, tile_dim1=256,
    //             tensor_dim0_stride=128, tensor_dim1_stride=32768
    v8i g1 = { (int)((2u << 16) | (1u << 20) | (6u << 22)),
               (int)(128u << 16),            // tensor_dim0[15:0] @ bits [63:48]
               (int)(256u << 16),            // tensor_dim1[15:0] @ bits [95:80]
               (int)(128u << 16),            // tile_dim0 @ bits [127:112]
               (int)256,                     // tile_dim1 @ bits [143:128]
               (int)128,                     // tensor_dim0_stride[31:0]
               (int)0x80000000u,             // tensor_dim1_stride[15:0]=32768 @ [223:208]
               0 };
    v4i gz = { 0, 0, 0, 0 };
#if __clang_major__ >= 23
    v8i gz8 = { 0, 0, 0, 0, 0, 0, 0, 0 };
    __builtin_amdgcn_tensor_load_to_lds(g0, g1, gz, gz, gz8, 0);
#else
    __builtin_amdgcn_tensor_load_to_lds(g0, g1, gz, gz, 0);
#endif
    __builtin_amdgcn_s_wait_tensorcnt(0);    // issuing wave sees completion ...
  }
#else
  {
    const float4* src = reinterpret_cast<const float4*>(Aout + (size_t)rowBase * HH);
    for (int idx = tid; idx < NN * HH / 4; idx += 256) {
      float4 vv = src[idx];
      int r  = idx >> 5;
      int hh = (idx & 31) * 4;
      float* dst = As + r * ASTR + hh;
      dst[0] = vv.x; dst[1] = vv.y; dst[2] = vv.z; dst[3] = vv.w;
    }
  }
#endif
  for (int idx = tid; idx < IBLK * HH; idx += 256)
    Bs[idx] = Bout[(size_t)(rowBase + i0 + (idx >> 7)) * HH + (idx & 127)];
  if (tid < HH) { wds[tid] = W1[2 * HH * HH + tid]; w2s[tid] = W2[tid]; }
  if (tid < IBLK * 3) pis[tid] = pos[(rowBase + i0 + tid / 3) * 3 + tid % 3];
  __syncthreads();                           // ... barrier publishes LDS to all waves

  const int j = tid;                         // this thread owns column j of the pair matrix
  const float pj0 = pos[(rowBase + j) * 3 + 0];
  const float pj1 = pos[(rowBase + j) * 3 + 1];
  const float pj2 = pos[(rowBase + j) * 3 + 2];

  float dist[IBLK];
  #pragma unroll
  for (int ii = 0; ii < IBLK; ++ii) {
    float dx = pj0 - pis[ii * 3 + 0];
    float dy = pj1 - pis[ii * 3 + 1];
    float dz = pj2 - pis[ii * 3 + 2];
    float sq = dx * dx + dy * dy + dz * dz;
    dist[ii] = sq > 0.f ? sqrtf(sq) : 0.f;   // safe_norm
  }

  float acc[IBLK];
  #pragma unroll
  for (int ii = 0; ii < IBLK; ++ii) acc[ii] = 0.f;

  const float L2E = 1.44269504088896340736f;
  for (int hh = 0; hh < HH; ++hh) {
    const float a  = As[j * ASTR + hh];      // conflict-free indexed read
    const float wd = wds[hh];                // broadcast
    const float w2 = w2s[hh];                // broadcast
    #pragma unroll
    for (int ii = 0; ii < IBLK; ++ii) {
      float pre = a + Bs[ii * HH + hh] + dist[ii] * wd;  // b1 already folded into Bs
      float sig = __builtin_amdgcn_rcpf(1.f + __builtin_amdgcn_exp2f(-L2E * pre));
      acc[ii] += pre * sig * w2;             // silu(pre) * W2[h]
    }
  }

  const float b2v   = b2[0];
  const int  waveId = tid >> 5;
  #pragma unroll
  for (int ii = 0; ii < IBLK; ++ii) {
    const float coeff = acc[ii] + b2v;
    float c0 = coeff * (pj0 - pis[ii * 3 + 0]);
    float c1 = coeff * (pj1 - pis[ii * 3 + 1]);
    float c2 = coeff * (pj2 - pis[ii * 3 + 2]);
    #pragma unroll
    for (int off = 16; off >= 1; off >>= 1) {   // wave32 butterfly reduction
      c0 += __shfl_xor(c0, off, 32);
      c1 += __shfl_xor(c1, off, 32);
      c2 += __shfl_xor(c2, off, 32);
    }
    if ((tid & 31) == 0) {
      red[waveId * (IBLK * 3) + ii * 3 + 0] = c0;
      red[waveId * (IBLK * 3) + ii * 3 + 1] = c1;
      red[waveId * (IBLK * 3) + ii * 3 + 2] = c2;
    }
  }
  __syncthreads();
  if (tid < IBLK * 3) {                      // fixed-order final sum: deterministic
    float s = 0.f;
    #pragma unroll
    for (int w = 0; w < 8; ++w) s += red[w * (IBLK * 3) + tid];
    out[(rowBase + i0 + tid / 3) * 3 + tid % 3] = s;
  }
}

extern "C" void kernel_launch(void* const* d_in, const int* in_sizes, int n_in,
                              void* d_out, int out_size, void* d_ws, size_t ws_size,
                              hipStream_t stream) {
  const float* hmat = (const float*)d_in[0];   // (2048,128)
  const float* pos  = (const float*)d_in[1];   // (2048,3)
  // d_in[2] = batch (int64): implied by layout, unused
  const float* W1   = (const float*)d_in[3];   // (257,128)
  const float* b1   = (const float*)d_in[4];   // (128,)
  const float* W2   = (const float*)d_in[5];   // (128,1)
  const float* b2   = (const float*)d_in[6];   // (1,)
  float* out  = (float*)d_out;                 // (2048,3)
  float* Aout = (float*)d_ws;                  // 2048*128 f32 = 1 MB
  float* Bout = Aout + (size_t)BB * NN * HH;   // 2048*128 f32 = 1 MB (Bm + b1)

  // Phase 1: one 16x16 WMMA tile per (single-wave) workgroup; K=128 -> 32 WMMAs.
  evh_gemm_wmma<<<2048, 32, 0, stream>>>(hmat, W1, b1, Aout, Bout);

  // Phase 2: one WG per (batch, 8-row i-block). ~135 KB dynamic LDS (<320 KB/WGP).
  const size_t lds = (size_t)(NN * (HH + 1) + 8 * HH + HH + HH + 24 + 8 * 24) * sizeof(float);
  evh_pair_kernel<<<dim3(BB * (NN / 8)), 256, lds, stream>>>(pos, Aout, Bout, W1, W2, b2, out);
}